// GIADA_L4P1_66262755442983
// MI455X (gfx1250) — compile-verified
//
#include <hip/hip_runtime.h>
#include <hip/hip_bf16.h>
#include <stdint.h>

#define BATCH   64
#define NB      45
#define DSYN    28
#define NCH     10
#define DM      20
#define APICALN 30
#define W4COLS  19    // napical(15) + G0(4)
#define MAXS    1280  // padded X-row staging size (floats); harness S = 1278
#define NF      10    // async b64 chunks per lane per row: NF*64*8B = 5120B >= 5112B

// ---------------- async-to-LDS (probe-confirmed on this toolchain) ----------------
#if __has_builtin(__builtin_amdgcn_global_load_async_to_lds_b64)
#define HAVE_ASYNC_LDS 1
#endif
#if __has_builtin(__builtin_amdgcn_s_wait_asynccnt)
#define HAVE_WAIT_ASYNC 1
#endif

// Builtin signature (decoded from clang diagnostic, addrspaces printed as CUDA quals):
//   void __builtin_amdgcn_global_load_async_to_lds_b64(
//       int __attribute__((vector_size(8))) __device__(as1)* src,
//       int __attribute__((vector_size(8))) __shared__(as3)* dst,
//       imm int offset, imm int cpol)
typedef int v2i32 __attribute__((vector_size(8)));
typedef __attribute__((address_space(1))) v2i32* as1_v2i;
typedef __attribute__((address_space(3))) v2i32* as3_v2i;

template <int N>
__device__ __forceinline__ void wait_async() {
#if defined(HAVE_ASYNC_LDS)
#if defined(HAVE_WAIT_ASYNC)
    __builtin_amdgcn_s_wait_asynccnt(N);
#else
    asm volatile("s_wait_asynccnt %0" ::"i"(N) : "memory");
#endif
#endif
}

// ---------------- math helpers ----------------
__device__ __forceinline__ float softplusf(float x) {
    return (x > 20.0f) ? x : log1pf(__expf(x));
}
__device__ __forceinline__ float sigmf(float x) { return 1.0f / (1.0f + __expf(-x)); }
__device__ __forceinline__ float tanh_f(float x) { return 2.0f * sigmf(2.0f * x) - 1.0f; }

typedef float v2f __attribute__((ext_vector_type(2)));
typedef float v8f __attribute__((ext_vector_type(8)));

// =====================================================================================
// Kernel A1: chunk-local synapse scan.  Grid: BATCH*NCH blocks, 64 threads.
// X row (5112 B) is staged into LDS with double-buffered async b64 loads: all 64 lanes
// issue a fixed NF async ops per row (padding chunk clamps its global address into the
// row), so "s_wait_asynccnt <= NF" after issuing row t+1 guarantees row t is resident.
// Lanes 0..44 then gather their branch's 28 synapses from LDS (float4 fast path when
// the indices are contiguous & 16B aligned) and advance the private scan state.
// =====================================================================================
__global__ void __launch_bounds__(64)
k_scan_local(const float* __restrict__ X,
             const float* __restrict__ log_tau_s,
             const float* __restrict__ w_s_raw,
             const int*   __restrict__ bidx,
             float* __restrict__ branches,   // [B,T,45]
             float* __restrict__ sEnd,       // [B,NCH,1260]
             int T, int L, int S) {
    const int b  = blockIdx.x / NCH;
    const int c  = blockIdx.x % NCH;
    const int t0 = c * L;
    const int t1 = (t0 + L < T) ? (t0 + L) : T;
    if (t0 >= t1) return;                       // block-uniform
    const int tid = threadIdx.x;
    const int j   = tid;

    __shared__ float xbuf[2][MAXS];
    const int n64 = S >> 1;                     // 8B chunks per row (S even)

    int   ids[DSYN];
    float kap[DSYN], wsy[DSYN], u[DSYN];
    bool  contig = false;
    if (j < NB) {
        contig = true;
#pragma unroll
        for (int k = 0; k < DSYN; ++k) {
            int id = bidx[j * DSYN + k];
            ids[k] = id;
            float ts = softplusf(log_tau_s[id]) + 0.001f;
            kap[k] = __expf(-1.0f / ts);
            wsy[k] = softplusf(w_s_raw[id]);
            u[k] = 0.0f;
            contig = contig && (id == ids[0] + k);
        }
        contig = contig && ((ids[0] & 3) == 0);
    }

    // ---- row staging ----
    auto fill_row = [&](int t, int bufi) {
        const char* gsrc = (const char*)(X + ((size_t)b * T + (size_t)t) * S);
        float*      ldst = &xbuf[bufi][0];
#pragma unroll
        for (int q = 0; q < NF; ++q) {
            int e  = q * 64 + tid;
            int ee = (e < n64) ? e : 0;         // clamp padding chunks into the row
#if defined(HAVE_ASYNC_LDS)
            __builtin_amdgcn_global_load_async_to_lds_b64(
                (as1_v2i)(uintptr_t)(gsrc + (size_t)ee * 8),
                (as3_v2i)(uint32_t)(uintptr_t)(ldst + e * 2),
                0, 0);
#else
            float2 v = *(const float2*)(gsrc + (size_t)ee * 8);
            *(float2*)(ldst + e * 2) = v;
#endif
        }
        if ((S & 1) && tid == 0)                // odd-S tail (not hit for S=1278)
            ldst[S - 1] = ((const float*)gsrc)[S - 1];
    };

    fill_row(t0, 0);
    for (int t = t0; t < t1; ++t) {
        const int  cur     = (t - t0) & 1;
        const bool hasNext = (t + 1 < t1);
        if (hasNext) {
            fill_row(t + 1, cur ^ 1);
            wait_async<NF>();                   // row t complete (in-order per wave)
        } else {
            wait_async<0>();
        }
        __syncthreads();                        // cross-wave LDS visibility

        if (j < NB) {
            float a0 = 0.f, a1 = 0.f, a2 = 0.f, a3 = 0.f;
            if (contig) {
                const float* xb = &xbuf[cur][ids[0]];
#pragma unroll
                for (int k = 0; k < DSYN; k += 4) {
                    float4 x4 = *(const float4*)(xb + k);
                    u[k + 0] = kap[k + 0] * u[k + 0] + wsy[k + 0] * x4.x;
                    u[k + 1] = kap[k + 1] * u[k + 1] + wsy[k + 1] * x4.y;
                    u[k + 2] = kap[k + 2] * u[k + 2] + wsy[k + 2] * x4.z;
                    u[k + 3] = kap[k + 3] * u[k + 3] + wsy[k + 3] * x4.w;
                    a0 += u[k + 0]; a1 += u[k + 1]; a2 += u[k + 2]; a3 += u[k + 3];
                }
            } else {
#pragma unroll
                for (int k = 0; k < DSYN; k += 4) {
                    float x0 = xbuf[cur][ids[k + 0]];
                    float x1 = xbuf[cur][ids[k + 1]];
                    float x2 = xbuf[cur][ids[k + 2]];
                    float x3 = xbuf[cur][ids[k + 3]];
                    u[k + 0] = kap[k + 0] * u[k + 0] + wsy[k + 0] * x0;
                    u[k + 1] = kap[k + 1] * u[k + 1] + wsy[k + 1] * x1;
                    u[k + 2] = kap[k + 2] * u[k + 2] + wsy[k + 2] * x2;
                    u[k + 3] = kap[k + 3] * u[k + 3] + wsy[k + 3] * x3;
                    a0 += u[k + 0]; a1 += u[k + 1]; a2 += u[k + 2]; a3 += u[k + 3];
                }
            }
            branches[((size_t)b * T + t) * NB + j] = (a0 + a1) + (a2 + a3);
        }
        __syncthreads();                        // readers done before buffer reuse
    }

    if (j < NB) {
        float* se = sEnd + ((size_t)b * NCH + c) * (NB * DSYN) + j * DSYN;
#pragma unroll
        for (int k = 0; k < DSYN; ++k) se[k] = u[k];
    }
}

// =====================================================================================
// Kernel A2: sequential chunk combine:  sStart[c+1] = kappa^L * sStart[c] + sEnd[c]
// =====================================================================================
__global__ void k_combine(const float* __restrict__ sEnd,
                          const int*   __restrict__ bidx,
                          const float* __restrict__ log_tau_s,
                          float* __restrict__ sStart,        // [B,NCH,1260]
                          int L) {
    int e = blockIdx.x * blockDim.x + threadIdx.x;
    const int PER = NB * DSYN;
    if (e >= BATCH * PER) return;
    int b = e / PER;
    int s = e % PER;
    float ts  = softplusf(log_tau_s[bidx[s]]) + 0.001f;
    float dec = __expf(-(float)L / ts);
    float ss = 0.0f;
    sStart[((size_t)b * NCH + 0) * PER + s] = 0.0f;
    for (int c = 0; c + 1 < NCH; ++c) {
        ss = dec * ss + sEnd[((size_t)b * NCH + c) * PER + s];
        sStart[((size_t)b * NCH + (c + 1)) * PER + s] = ss;
    }
}

// =====================================================================================
// Kernel A3: per-chunk correction:  branches[t] += sum_k kappa^(t-t0+1) * sStart.
// =====================================================================================
__global__ void k_correct(const float* __restrict__ sStart,
                          const int*   __restrict__ bidx,
                          const float* __restrict__ log_tau_s,
                          float* __restrict__ branches,
                          int T, int L) {
    const int b  = blockIdx.x / (NCH - 1);
    const int c  = 1 + blockIdx.x % (NCH - 1);
    const int t0 = c * L;
    const int t1 = (t0 + L < T) ? (t0 + L) : T;
    if (t0 >= t1) return;
    const int j = threadIdx.x;
    if (j >= NB) return;

    float kap[DSYN], u[DSYN];
    const float* ss = sStart + ((size_t)b * NCH + c) * (NB * DSYN) + j * DSYN;
#pragma unroll
    for (int k = 0; k < DSYN; ++k) {
        int id = bidx[j * DSYN + k];
        float ts = softplusf(log_tau_s[id]) + 0.001f;
        kap[k] = __expf(-1.0f / ts);
        u[k] = ss[k];
    }
    for (int t = t0; t < t1; ++t) {
        float a0 = 0.f, a1 = 0.f, a2 = 0.f, a3 = 0.f;
#pragma unroll
        for (int k = 0; k < DSYN; k += 4) {
            u[k + 0] *= kap[k + 0]; u[k + 1] *= kap[k + 1];
            u[k + 2] *= kap[k + 2]; u[k + 3] *= kap[k + 3];
            a0 += u[k + 0]; a1 += u[k + 1]; a2 += u[k + 2]; a3 += u[k + 3];
        }
        branches[((size_t)b * T + t) * NB + j] += (a0 + a1) + (a2 + a3);
    }
}

// =====================================================================================
// Kernel B: WMMA GEMM  [64000 x 45] @ [45 x 13] -> post-activation [64000 x 16]
//   cols 0-3  : delta_g1 = tanh(branches@W1^T + b1)
//   cols 4-9  : delta_g2 = 2*sigmoid(gain2*(branches@W2^T - th2)) - 1
//   cols 10-12: c4 = branches[:,30:45] @ W4[:, :15]^T
// V_WMMA_F32_16X16X4_F32, K padded 45->48 (12 k-steps), N padded 13->16.
// =====================================================================================
__global__ void k_gemm_wmma(const float* __restrict__ branches,
                            const float* __restrict__ W1, const float* __restrict__ b1,
                            const float* __restrict__ W2, const float* __restrict__ th2,
                            const float* __restrict__ lg2,
                            const float* __restrict__ W4,
                            float* __restrict__ pre16, int Mtotal) {
    __shared__ float wcat[16][48];
    const int tid = threadIdx.x;
    for (int i = tid; i < 16 * 48; i += blockDim.x) {
        int n = i / 48, k = i % 48;
        float v = 0.0f;
        if (k < NB) {
            if (n < 4)       v = W1[n * NB + k];
            else if (n < 10) v = W2[(n - 4) * NB + k];
            else if (n < 13) { if (k >= APICALN) v = W4[(n - 10) * W4COLS + (k - APICALN)]; }
        }
        wcat[n][k] = v;
    }
    __syncthreads();

    const int wave = tid >> 5;
    const int lane = tid & 31;
    const int rowbase = (blockIdx.x * (blockDim.x >> 5) + wave) * 16;
    if (rowbase >= Mtotal) return;       // whole-wave uniform: EXEC stays all-ones

    const int mlane = lane & 15;         // A: M / B,D: N
    const int khalf = lane >> 4;         // A,B: K-half / D: M-half
    const float* arow = branches + (size_t)(rowbase + mlane) * NB;

    v8f acc = {0.f, 0.f, 0.f, 0.f, 0.f, 0.f, 0.f, 0.f};
#pragma unroll
    for (int kk = 0; kk < 12; ++kk) {
        const int ka = kk * 4 + khalf * 2;
        v2f a, bm;
        a.x  = (ka     < NB) ? arow[ka]     : 0.0f;
        a.y  = (ka + 1 < NB) ? arow[ka + 1] : 0.0f;
        bm.x = wcat[mlane][ka];
        bm.y = wcat[mlane][ka + 1];
        acc = __builtin_amdgcn_wmma_f32_16x16x4_f32(false, a, false, bm,
                                                    (short)0, acc, false, false);
    }

    const int n = mlane;
    const float bb1 = (n < 4) ? b1[n] : 0.0f;
    const float t2  = (n >= 4 && n < 10) ? th2[n - 4] : 0.0f;
    const float g2  = (n >= 4 && n < 10) ? __expf(lg2[n - 4]) : 0.0f;
#pragma unroll
    for (int v = 0; v < 8; ++v) {
        const int r = rowbase + v + 8 * khalf;
        float val = acc[v];
        if (n < 4)        val = tanh_f(val + bb1);
        else if (n < 10)  val = 2.0f * sigmf(g2 * (val - t2)) - 1.0f;
        else if (n >= 13) val = 0.0f;
        pre16[(size_t)r * 16 + n] = val;
    }
}

// =====================================================================================
// Kernel C: the irreducibly-sequential m-recurrence.  One lane per batch row.
// =====================================================================================
__global__ void __launch_bounds__(64, 1)
k_serial(const float* __restrict__ pre16,
         const float* __restrict__ log_tau_m,
         const float* __restrict__ W3, const float* __restrict__ th3,
         const float* __restrict__ W4, const float* __restrict__ th4,
         const float* __restrict__ lg4,
         const float* __restrict__ W5, const float* __restrict__ b5,
         const float* __restrict__ Wy, const float* __restrict__ by,
         float* __restrict__ y, int T) {
    const int b = threadIdx.x;
    if (b >= BATCH) return;

    const int   gsz[5] = {4, 6, 5, 3, 2};
    const float blo[5] = {1.f, 8.f, 30.f, 60.f, 100.f};
    const float bhi[5] = {8.f, 35.f, 80.f, 150.f, 500.f};
    float kap[DM], lamk[DM];
    {
        int i = 0;
#pragma unroll
        for (int g = 0; g < 5; ++g) {
#pragma unroll 8
            for (int q = 0; q < gsz[g]; ++q) {
                float lt = log_tau_m[i];
                float lo = logf(blo[g]), hi = logf(bhi[g]);
                lt = fminf(fmaxf(lt, lo), hi);
                float tau = __expf(lt);
                kap[i]  = __expf(-1.0f / tau);
                lamk[i] = 10.0f * (1.0f - kap[i]);
                ++i;
            }
        }
    }

    float w3[5][6], t3[5], w4b[3][4], t4[3], g4[3], w5[2][4], bb5[2], wy[2][DM], bby[2];
#pragma unroll
    for (int r = 0; r < 5; ++r) {
        t3[r] = th3[r];
#pragma unroll
        for (int jj = 0; jj < 6; ++jj) w3[r][jj] = W3[r * 6 + jj];
    }
#pragma unroll
    for (int r = 0; r < 3; ++r) {
        t4[r] = th4[r];
        g4[r] = __expf(lg4[r]);
#pragma unroll
        for (int jj = 0; jj < 4; ++jj) w4b[r][jj] = W4[r * W4COLS + 15 + jj];
    }
#pragma unroll
    for (int r = 0; r < 2; ++r) {
        bb5[r] = b5[r];
        bby[r] = by[r];
#pragma unroll
        for (int jj = 0; jj < 4; ++jj) w5[r][jj] = W5[r * 4 + jj];
#pragma unroll
        for (int i = 0; i < DM; ++i) wy[r][i] = Wy[r * DM + i];
    }

    float m[DM];
#pragma unroll
    for (int i = 0; i < DM; ++i) m[i] = 0.0f;

    const float4* pr4 = (const float4*)(pre16 + (size_t)b * T * 16);
    float4 c0 = pr4[0], c1 = pr4[1], c2 = pr4[2], c3 = pr4[3];

    for (int t = 0; t < T; ++t) {
        float4 n0 = c0, n1 = c1, n2 = c2, n3 = c3;
        if (t + 1 < T) {
            n0 = pr4[(t + 1) * 4 + 0]; n1 = pr4[(t + 1) * 4 + 1];
            n2 = pr4[(t + 1) * 4 + 2]; n3 = pr4[(t + 1) * 4 + 3];
        }

        float p[16];
        *(float4*)&p[0] = c0; *(float4*)&p[4] = c1;
        *(float4*)&p[8] = c2; *(float4*)&p[12] = c3;

        float md[DM];
#pragma unroll
        for (int i = 0; i < DM; ++i) md[i] = kap[i] * m[i];

        float g1a[4], g2a[6];
#pragma unroll
        for (int jj = 0; jj < 4; ++jj) g1a[jj] = md[jj] + p[jj];
#pragma unroll
        for (int jj = 0; jj < 6; ++jj) g2a[jj] = md[4 + jj] + p[4 + jj];

        float dg[DM];
#pragma unroll
        for (int i = 0; i < 10; ++i) dg[i] = p[i];
#pragma unroll
        for (int r = 0; r < 5; ++r) {
            float a = -t3[r];
#pragma unroll
            for (int jj = 0; jj < 6; ++jj) a += g2a[jj] * w3[r][jj];
            dg[10 + r] = sigmf(a);
        }
#pragma unroll
        for (int r = 0; r < 3; ++r) {
            float a = p[10 + r] - t4[r];
#pragma unroll
            for (int jj = 0; jj < 4; ++jj) a += g1a[jj] * w4b[r][jj];
            dg[15 + r] = 2.0f * sigmf(g4[r] * a) - 1.0f;
        }
#pragma unroll
        for (int r = 0; r < 2; ++r) {
            float a = bb5[r];
#pragma unroll
            for (int jj = 0; jj < 4; ++jj) a += g1a[jj] * w5[r][jj];
            dg[18 + r] = tanh_f(a);
        }

        float y0 = bby[0], y1 = bby[1];
#pragma unroll
        for (int i = 0; i < DM; ++i) {
            m[i] = md[i] + lamk[i] * dg[i];
            y0 += m[i] * wy[0][i];
            y1 += m[i] * wy[1][i];
        }
        y[((size_t)b * T + t) * 2 + 0] = y0;
        y[((size_t)b * T + t) * 2 + 1] = y1;

        c0 = n0; c1 = n1; c2 = n2; c3 = n3;
    }
}

// =====================================================================================
// Host launch
// =====================================================================================
extern "C" void kernel_launch(void* const* d_in, const int* in_sizes, int n_in,
                              void* d_out, int out_size, void* d_ws, size_t ws_size,
                              hipStream_t stream) {
    const float* X         = (const float*)d_in[0];
    const float* log_tau_s = (const float*)d_in[1];
    const float* w_s_raw   = (const float*)d_in[2];
    const float* log_tau_m = (const float*)d_in[3];
    const float* W1        = (const float*)d_in[4];
    const float* b1        = (const float*)d_in[5];
    const float* W2        = (const float*)d_in[6];
    const float* th2       = (const float*)d_in[7];
    const float* lg2       = (const float*)d_in[8];
    const float* W3        = (const float*)d_in[9];
    const float* th3       = (const float*)d_in[10];
    const float* W4        = (const float*)d_in[11];
    const float* th4       = (const float*)d_in[12];
    const float* lg4       = (const float*)d_in[13];
    const float* W5        = (const float*)d_in[14];
    const float* b5        = (const float*)d_in[15];
    const float* Wy        = (const float*)d_in[16];
    const float* by        = (const float*)d_in[17];
    const int*   bidx      = (const int*)d_in[18];

    const int S = in_sizes[1];                       // 1278
    const int T = in_sizes[0] / (BATCH * S);         // 1000
    const int L = (T + NCH - 1) / NCH;               // 100

    float* ws       = (float*)d_ws;
    float* branches = ws;                                        // B*T*45
    float* pre16    = branches + (size_t)BATCH * T * NB;         // B*T*16
    float* sEnd     = pre16 + (size_t)BATCH * T * 16;            // B*NCH*1260
    float* sStart   = sEnd + (size_t)BATCH * NCH * NB * DSYN;    // B*NCH*1260

    // A1: parallel chunk-local scan with async double-buffered X staging
    k_scan_local<<<BATCH * NCH, 64, 0, stream>>>(X, log_tau_s, w_s_raw, bidx,
                                                 branches, sEnd, T, L, S);
    // A2: sequential chunk combine (tiny)
    {
        int tot = BATCH * NB * DSYN;
        k_combine<<<(tot + 255) / 256, 256, 0, stream>>>(sEnd, bidx, log_tau_s, sStart, L);
    }
    // A3: per-chunk exact correction of branch sums
    k_correct<<<BATCH * (NCH - 1), 64, 0, stream>>>(sStart, bidx, log_tau_s,
                                                    branches, T, L);
    // B: WMMA GEMM -> precomputed activations (dg1, dg2, c4)
    {
        int Mtot = BATCH * T;
        int rowsPerBlock = 16 * (256 / 32);          // 128
        int blocks = (Mtot + rowsPerBlock - 1) / rowsPerBlock;
        k_gemm_wmma<<<blocks, 256, 0, stream>>>(branches, W1, b1, W2, th2, lg2, W4,
                                                pre16, Mtot);
    }
    // C: sequential m-recurrence (lane = batch row)
    k_serial<<<1, 64, 0, stream>>>(pre16, log_tau_m, W3, th3, W4, th4, lg4,
                                   W5, b5, Wy, by, (float*)d_out, T);
}